// decompress_jpeg_18021682774939
// MI455X (gfx1250) — compile-verified
//
#include <hip/hip_runtime.h>

// CDNA5 / gfx1250 fused JPEG decode:
//   dequant -> 8x8 IDCT via V_WMMA_F32_16X16X4_F32 -> merge -> chroma 2x upsample
//   -> YCbCr->RGB -> clip, single pass. Memory floor: 96 MB in + 192 MB out =
//   288 MB @ 23.3 TB/s ~= 12.4 us; this revision cuts per-wave VALU (~1550 ->
//   ~330 instr/tile) so the kernel sits on the HBM roofline, not the VALU one.
//
// One wave32 per 16x16 output pixel tile, 4 tiles per wave (amortizes setup):
//   luma:   4 8x8 blocks packed as 16x16 X; D = Cd^T * X * Cd, Cd = diag(C', C')
//   chroma: Cb|Cr packed side-by-side in top half of a 16x16 X.
//   C'[x][u] = 0.5*alpha(x)*cos((2u+1)x*pi/16) -- the IDCT's 0.25 and alpha
//   weights folded into the constant matrix (table below), so the load stage
//   multiplies by the quant table only.
// Each 16x16x16 GEMM = 4 chained wmma_f32_16x16x4_f32 (K-chunks of 4).
//
// WMMA f32 16x16x4 operand layout (mirrors documented 16-bit layouts):
//   A (16x4): VGPR p, lane l -> (M = l&15, K = 2*(l>>4) + p)
//   B (4x16): VGPR p, lane l -> (K = 2*(l>>4) + p, N = l&15)
//   C/D:      VGPR i, lane l -> (M = i + 8*(l>>4), N = l&15)

typedef float v2f __attribute__((ext_vector_type(2)));
typedef float v4f __attribute__((ext_vector_type(4)));
typedef float v8f __attribute__((ext_vector_type(8)));

// C'[x][u] = 0.5*alpha(x)*cos((2u+1)*x*pi/16), row-major [x][u]
__device__ __constant__ float CP[64] = {
    0.35355339059327373f,  0.35355339059327373f,  0.35355339059327373f,  0.35355339059327373f,
    0.35355339059327373f,  0.35355339059327373f,  0.35355339059327373f,  0.35355339059327373f,
    0.49039264020161522f,  0.41573480615127262f,  0.27778511650980114f,  0.09754516100806412f,
   -0.09754516100806412f, -0.27778511650980114f, -0.41573480615127262f, -0.49039264020161522f,
    0.46193976625564337f,  0.19134171618254492f, -0.19134171618254492f, -0.46193976625564337f,
   -0.46193976625564337f, -0.19134171618254492f,  0.19134171618254492f,  0.46193976625564337f,
    0.41573480615127262f, -0.09754516100806412f, -0.49039264020161522f, -0.27778511650980114f,
    0.27778511650980114f,  0.49039264020161522f,  0.09754516100806412f, -0.41573480615127262f,
    0.35355339059327379f, -0.35355339059327379f, -0.35355339059327379f,  0.35355339059327379f,
    0.35355339059327379f, -0.35355339059327379f, -0.35355339059327379f,  0.35355339059327379f,
    0.27778511650980114f, -0.49039264020161522f,  0.09754516100806412f,  0.41573480615127262f,
   -0.41573480615127262f, -0.09754516100806412f,  0.49039264020161522f, -0.27778511650980114f,
    0.19134171618254492f, -0.46193976625564337f,  0.46193976625564337f, -0.19134171618254492f,
   -0.19134171618254492f,  0.46193976625564337f, -0.46193976625564337f,  0.19134171618254492f,
    0.09754516100806412f, -0.27778511650980114f,  0.41573480615127262f, -0.49039264020161522f,
    0.49039264020161522f, -0.41573480615127262f,  0.27778511650980114f, -0.09754516100806412f,
};

#define ROWP 20  // padded LDS row stride (floats): 16B-aligned v4 stores, conflict-free

__device__ __forceinline__ v8f wmma4(v2f a, v2f b, v8f c) {
    return __builtin_amdgcn_wmma_f32_16x16x4_f32(
        /*neg_a=*/false, a, /*neg_b=*/false, b,
        /*c_mod=*/(short)0, c, /*reuse_a=*/false, /*reuse_b=*/false);
}

// D = Cd^T * X * Cd + cinit*J, X in stg (destroyed: reused for T = X*Cd).
__device__ __forceinline__ v8f tile_idct(float (*stg)[ROWP], const float* cd,
                                         int lo, int hi, float cinit) {
    v8f acc = {};
#pragma unroll
    for (int kc = 0; kc < 4; ++kc) {
        v2f a, b;
        a.x = stg[lo][4 * kc + 2 * hi + 0];   // A = X
        a.y = stg[lo][4 * kc + 2 * hi + 1];
        b.x = cd[2 * kc + 0];                 // B = Cd
        b.y = cd[2 * kc + 1];
        acc = wmma4(a, b, acc);
    }
    __syncthreads();
#pragma unroll
    for (int i = 0; i < 8; ++i)               // park T (D-layout -> LDS)
        stg[i + 8 * hi][lo] = acc[i];
    __syncthreads();
    v8f acc2 = {cinit, cinit, cinit, cinit, cinit, cinit, cinit, cinit};
#pragma unroll
    for (int kc = 0; kc < 4; ++kc) {
        v2f a, b;
        a.x = cd[2 * kc + 0];                 // A = Cd^T (same per-lane values)
        a.y = cd[2 * kc + 1];
        b.x = stg[4 * kc + 2 * hi + 0][lo];   // B = T
        b.y = stg[4 * kc + 2 * hi + 1][lo];
        acc2 = wmma4(a, b, acc2);
    }
    return acc2;
}

__global__ __launch_bounds__(32)
void jpeg_decode_tiles(const float* __restrict__ ycoef,
                       const float* __restrict__ cbcoef,
                       const float* __restrict__ crcoef,
                       const float* __restrict__ ytab,
                       const float* __restrict__ ctab,
                       float* __restrict__ out) {
    constexpr int H = 1024, W = 1024, HW = H * W;
    constexpr int YBPR = W / 8;               // 128 luma blocks per row
    constexpr int CBPR = (W / 2) / 8;         // 64 chroma blocks per row
    constexpr int YBLKS = (H / 8) * YBPR;     // 16384 per batch
    constexpr int CBLKS = (H / 16) * CBPR;    // 4096 per batch
    constexpr int TPW = 4;                    // tiles per wave

    const int tc0 = blockIdx.x * TPW, tr = blockIdx.y, b = blockIdx.z;
    const int lane = threadIdx.x & 31;
    const int hi = lane >> 4, lo = lane & 15;

    __shared__ float stg[16][ROWP];   // X / T staging
    __shared__ float chr[16][ROWP];   // chroma spatial: cb cols 0-7, cr cols 8-15

    // Per-lane Cd operand values (serve as both GEMM1-B and GEMM2-A).
    float cd[8];
#pragma unroll
    for (int kc = 0; kc < 4; ++kc) {
#pragma unroll
        for (int p = 0; p < 2; ++p) {
            const int K = 4 * kc + 2 * hi + p;
            const float v = CP[((K & 7) << 3) | (lo & 7)];
            cd[2 * kc + p] = ((K ^ lo) & 8) ? 0.0f : v;   // block-diagonal
        }
    }

    const v4f* yc4 = (const v4f*)ycoef;
    const v4f* cb4 = (const v4f*)cbcoef;
    const v4f* cr4 = (const v4f*)crcoef;
    const v4f* yt4 = (const v4f*)ytab;
    const v4f* ct4 = (const v4f*)ctab;

    for (int t = 0; t < TPW; ++t) {
        const int tc = tc0 + t;

        // ---- chroma: load Cb|Cr packed (one v4 per lane), dequant ----
        {
            const int comp = lane >> 4;            // 0 = Cb, 1 = Cr
            const int ql = lane & 15;              // float4 index within block
            const int x = ql >> 1, yy = (ql & 1) * 4;
            const v4f* src = comp ? cr4 : cb4;
            const v4f val =
                src[(size_t)(b * CBLKS + tr * CBPR + tc) * 16 + ql] * ct4[ql];
            *(v4f*)&stg[x][8 * comp + yy] = val;
            *(v4f*)&stg[8 + (lane >> 2)][(lane & 3) * 4] = (v4f){0.f, 0.f, 0.f, 0.f};
        }
        __syncthreads();
        {
            const v8f dc = tile_idct(stg, cd, lo, hi, 0.0f);
            __syncthreads();
#pragma unroll
            for (int i = 0; i < 8; ++i)
                chr[i + 8 * hi][lo] = dc[i];
        }
        __syncthreads();

        // ---- luma: load 2x2 blocks as 16x16 (two v4 per lane), dequant ----
#pragma unroll
        for (int k2 = 0; k2 < 2; ++k2) {
            const int q = lane + 32 * k2;          // 0..63 float4s
            const int blk = q >> 4;                // 0..3 -> (bi, bj)
            const int ql = q & 15;
            const int x = ql >> 1, yy = (ql & 1) * 4;
            const int bi = blk >> 1, bj = blk & 1;
            const int n = (2 * tr + bi) * YBPR + (2 * tc + bj);
            const v4f val = yc4[(size_t)(b * YBLKS + n) * 16 + ql] * yt4[ql];
            *(v4f*)&stg[8 * bi + x][8 * bj + yy] = val;
        }
        __syncthreads();

        const v8f dy = tile_idct(stg, cd, lo, hi, 128.0f);  // +128 folded into C

        // ---- combine: upsample chroma, YCbCr->RGB, clip, store ----
        const size_t obase = (size_t)b * 3 * HW + (size_t)(tr * 16) * W + tc * 16 + lo;
#pragma unroll
        for (int ip = 0; ip < 4; ++ip) {           // chroma row pair (2x upsample)
            const int uc = ip + 4 * hi;
            const float cbv = chr[uc][lo >> 1];
            const float crv = chr[uc][8 + (lo >> 1)];
            const float rt = 1.402f * crv;
            const float gt = -0.344136f * cbv - 0.714136f * crv;
            const float bt = 1.772f * cbv;
#pragma unroll
            for (int s = 0; s < 2; ++s) {
                const int i = 2 * ip + s;
                const float Yv = dy[i];            // already includes +128
                float R = Yv + rt, G = Yv + gt, Bv = Yv + bt;
                R = fminf(fmaxf(R, 0.0f), 255.0f);
                G = fminf(fmaxf(G, 0.0f), 255.0f);
                Bv = fminf(fmaxf(Bv, 0.0f), 255.0f);
                const size_t p = obase + (size_t)(i + 8 * hi) * W;
                out[p] = R;
                out[p + HW] = G;
                out[p + 2 * (size_t)HW] = Bv;
            }
        }
        __syncthreads();
    }
}

extern "C" void kernel_launch(void* const* d_in, const int* in_sizes, int n_in,
                              void* d_out, int out_size, void* d_ws, size_t ws_size,
                              hipStream_t stream) {
    const float* y  = (const float*)d_in[0];
    const float* cb = (const float*)d_in[1];
    const float* cr = (const float*)d_in[2];
    const float* yt = (const float*)d_in[3];
    const float* ct = (const float*)d_in[4];
    float* out = (float*)d_out;

    const int batches = in_sizes[0] / (128 * 128 * 64);   // B = 16
    dim3 grid((1024 / 16) / 4, 1024 / 16, batches);       // 4 tiles per wave
    jpeg_decode_tiles<<<grid, dim3(32, 1, 1), 0, stream>>>(y, cb, cr, yt, ct, out);
}